// GAT_60464549593448
// MI455X (gfx1250) — compile-verified
//
#include <hip/hip_runtime.h>
#include <math.h>

typedef __attribute__((ext_vector_type(2))) float v2f;
typedef __attribute__((ext_vector_type(8))) float v8f;

#define LN_EPS 1e-5f
#define NEG_SLOPE 0.2f

// ---------------------------------------------------------------------------
// GEMM via V_WMMA_F32_16X16X4_F32.  C[M,Nc] = A[M,K] @ B[K,Nc] (+ bias[Nc]).
// One wave per 16x16 output tile; K stepped by 4, unrolled x2.
// A-tile per lane: row = tm+(lane&15), K = k+2*(lane>>4) .. +1
// B-tile per lane: col = tn+(lane&15), same K pair.
// C: VGPR j -> row tm + j + 8*(lane>>4), col tn+(lane&15).
// Partial M tiles handled branchlessly: clamp row, multiply A by 0/1 mask.
// ---------------------------------------------------------------------------
__global__ void __launch_bounds__(256) k_gemm_wmma(
    const float* __restrict__ A, const float* __restrict__ B,
    const float* __restrict__ bias, float* __restrict__ C,
    int M, int Nc, int K)
{
  long wave = ((long)blockIdx.x * blockDim.x + threadIdx.x) >> 5;
  int lane  = threadIdx.x & 31;
  int tilesN = Nc >> 4;
  int tilesM = (M + 15) >> 4;
  if (wave >= (long)tilesM * tilesN) return;
  int tm = (int)(wave / tilesN) << 4;
  int tn = (int)(wave % tilesN) << 4;
  int r   = lane & 15;
  int kk  = (lane >> 4) << 1;     // 0 or 2
  int row = tm + r;
  int col = tn + r;
  float mask = (row < M) ? 1.0f : 0.0f;   // branchless partial-tile guard
  int rowc = row < M ? row : (M - 1);     // clamped: always a valid row
  const float* Ap = A + (size_t)rowc * K + kk;
  const float* Bp = B + (size_t)kk * Nc + col;
  const size_t bstep = (size_t)4 * Nc;
  v8f acc = {};
  #pragma unroll 2
  for (int k = 0; k < K; k += 4) {
    v2f a = *(const v2f*)Ap;    // 8B aligned: kk even, K multiple of 4
    a.x *= mask; a.y *= mask;
    v2f b;
    b.x = Bp[0];
    b.y = Bp[Nc];
    acc = __builtin_amdgcn_wmma_f32_16x16x4_f32(
        /*neg_a=*/false, a, /*neg_b=*/false, b,
        /*c_mod=*/(short)0, acc, /*reuse_a=*/false, /*reuse_b=*/false);
    Ap += 4;
    Bp += bstep;
  }
  float bv = bias ? bias[col] : 0.0f;
  int mhi = (lane >> 4) << 3;
  #pragma unroll
  for (int j = 0; j < 8; ++j) {
    int orow = tm + mhi + j;
    if (orow < M) C[(size_t)orow * Nc + col] = acc[j] + bv;
  }
}

// ---------------------------------------------------------------------------
__global__ void __launch_bounds__(256) k_fill_zero(float* p, long n) {
  long t = (long)blockIdx.x * blockDim.x + threadIdx.x;
  if (t < n) p[t] = 0.0f;
}

__global__ void __launch_bounds__(256) k_fill_bias(
    float* __restrict__ out, const float* __restrict__ b, long n, int D) {
  long t = (long)blockIdx.x * blockDim.x + threadIdx.x;
  if (t < n * (long)D) out[t] = b[t % D];
}

__global__ void __launch_bounds__(256) k_copy(
    const float* __restrict__ s, float* __restrict__ d, long n) {
  long t = (long)blockIdx.x * blockDim.x + threadIdx.x;
  if (t < n) d[t] = s[t];
}

// asrc[n,H] = sum_c h[n,h,c]*a_src[h,c]; adst likewise. thread per (node,head)
__global__ void __launch_bounds__(256) k_attn_coef(
    const float* __restrict__ h, const float* __restrict__ a_s,
    const float* __restrict__ a_d, float* __restrict__ asrc,
    float* __restrict__ adst, long n, int H, int C)
{
  long t = (long)blockIdx.x * blockDim.x + threadIdx.x;
  if (t >= n * H) return;
  long node = t / H; int hd = (int)(t - node * H);
  const float* hp = h + node * (long)(H * C) + (long)hd * C;
  const float* as = a_s + hd * C;
  const float* ad = a_d + hd * C;
  float s1 = 0.0f, s2 = 0.0f;
  for (int c = 0; c < C; ++c) { float v = hp[c]; s1 += v * as[c]; s2 += v * ad[c]; }
  asrc[t] = s1; adst[t] = s2;
}

// softmax denominator: z[dst,h] += exp(leaky(asrc[src,h]+adst[dst,h]))
// edges e<E from edge_index, e>=E are self-loops (e-E, e-E). thread per (edge,head)
__global__ void __launch_bounds__(256) k_edge_denom(
    const int* __restrict__ ei, long E, long n, int H,
    const float* __restrict__ asrc, const float* __restrict__ adst,
    float* __restrict__ z)
{
  long t = (long)blockIdx.x * blockDim.x + threadIdx.x;
  if (t >= (E + n) * H) return;
  long e = t / H; int hd = (int)(t - e * H);
  int s, d;
  if (e < E) { s = ei[e]; d = ei[E + e]; } else { s = d = (int)(e - E); }
  float x = asrc[(long)s * H + hd] + adst[(long)d * H + hd];
  x = x > 0.0f ? x : NEG_SLOPE * x;
  atomicAdd(&z[(long)d * H + hd], expf(x));
}

// out[dst,:] += alpha * h[src,:]   one wave per edge, NCH floats per lane
template <int NCH, int CC, int HH>
__global__ void __launch_bounds__(256) k_edge_aggregate(
    const int* __restrict__ ei, long E, long n,
    const float* __restrict__ h, const float* __restrict__ asrc,
    const float* __restrict__ adst, const float* __restrict__ z,
    float* __restrict__ out)
{
  const int D = NCH * 32;
  long wave = ((long)blockIdx.x * blockDim.x + threadIdx.x) >> 5;
  int lane  = threadIdx.x & 31;
  if (wave >= E + n) return;
  int s, d;
  if (wave < E) { s = ei[wave]; d = ei[E + wave]; } else { s = d = (int)(wave - E); }
  #pragma unroll
  for (int j = 0; j < NCH; ++j) {
    int c  = j * 32 + lane;
    int hd = c / CC;
    float x = asrc[(long)s * HH + hd] + adst[(long)d * HH + hd];
    x = x > 0.0f ? x : NEG_SLOPE * x;
    float alpha = expf(x) / z[(long)d * HH + hd];
    atomicAdd(&out[(size_t)d * D + c], alpha * h[(size_t)s * D + c]);
  }
}

// adjacency-mean accumulation for dual_fuse: thread per (group-edge, feature)
__global__ void __launch_bounds__(256) k_adj_accum(
    const int* __restrict__ eg, long EG,
    const float* __restrict__ gout, float* __restrict__ adj,
    float* __restrict__ cnt, int D)
{
  long t = (long)blockIdx.x * blockDim.x + threadIdx.x;
  if (t >= EG * D) return;
  long e = t / D; int dd = (int)(t - e * D);
  int ctr = eg[e];        // edge_index2[0]
  int nb  = eg[EG + e];   // edge_index2[1]
  atomicAdd(&adj[(size_t)ctr * D + dd], gout[(size_t)nb * D + dd]);
  if (dd == 0) atomicAdd(&cnt[ctr], 1.0f);
}

__global__ void __launch_bounds__(256) k_adj_finalize(
    float* __restrict__ adj, const float* __restrict__ cnt, long G, int D) {
  long t = (long)blockIdx.x * blockDim.x + threadIdx.x;
  if (t >= G * D) return;
  adj[t] /= fmaxf(cnt[t / D], 1.0f);
}

// upd = x + (x.gf)*gf + (x.af)*af   one wave per node
template <int NCH>
__global__ void __launch_bounds__(256) k_fuse(
    const float* __restrict__ xout, const float* __restrict__ gout,
    const float* __restrict__ adj, const int* __restrict__ gidx,
    float* __restrict__ fused, long n)
{
  const int D = NCH * 32;
  long wave = ((long)blockIdx.x * blockDim.x + threadIdx.x) >> 5;
  int lane  = threadIdx.x & 31;
  if (wave >= n) return;
  int g = gidx[wave];
  const float* xo = xout + (size_t)wave * D;
  const float* gf = gout + (size_t)g * D;
  const float* af = adj  + (size_t)g * D;
  float xv[NCH], gv[NCH], av[NCH];
  float ig = 0.0f, ia = 0.0f;
  #pragma unroll
  for (int j = 0; j < NCH; ++j) {
    int c = j * 32 + lane;
    xv[j] = xo[c]; gv[j] = gf[c]; av[j] = af[c];
    ig += xv[j] * gv[j]; ia += xv[j] * av[j];
  }
  #pragma unroll
  for (int off = 16; off > 0; off >>= 1) {
    ig += __shfl_xor(ig, off, 32);
    ia += __shfl_xor(ia, off, 32);
  }
  float* fo = fused + (size_t)wave * D;
  #pragma unroll
  for (int j = 0; j < NCH; ++j) {
    int c = j * 32 + lane;
    fo[c] = xv[j] + ig * gv[j] + ia * av[j];
  }
}

// LayerNorm (+optional ELU), one wave per row
template <int NCH>
__global__ void __launch_bounds__(256) k_layernorm(
    const float* __restrict__ in, const float* __restrict__ gw,
    const float* __restrict__ bw, float* __restrict__ out, long n, int do_elu)
{
  const int D = NCH * 32;
  long wave = ((long)blockIdx.x * blockDim.x + threadIdx.x) >> 5;
  int lane  = threadIdx.x & 31;
  if (wave >= n) return;
  const float* xi = in + (size_t)wave * D;
  float v[NCH]; float s = 0.0f;
  #pragma unroll
  for (int j = 0; j < NCH; ++j) { v[j] = xi[j * 32 + lane]; s += v[j]; }
  #pragma unroll
  for (int off = 16; off > 0; off >>= 1) s += __shfl_xor(s, off, 32);
  float mu = s / (float)D;
  float q = 0.0f;
  #pragma unroll
  for (int j = 0; j < NCH; ++j) { float dd = v[j] - mu; q += dd * dd; }
  #pragma unroll
  for (int off = 16; off > 0; off >>= 1) q += __shfl_xor(q, off, 32);
  float inv = 1.0f / sqrtf(q / (float)D + LN_EPS);
  float* oo = out + (size_t)wave * D;
  #pragma unroll
  for (int j = 0; j < NCH; ++j) {
    int c = j * 32 + lane;
    float y = (v[j] - mu) * inv * gw[c] + bw[c];
    if (do_elu) y = y > 0.0f ? y : expm1f(y);
    oo[c] = y;
  }
}

// ---------------------------------------------------------------------------
static inline dim3 grid1(long threads) {
  return dim3((unsigned)((threads + 255) / 256));
}

extern "C" void kernel_launch(void* const* d_in, const int* in_sizes, int n_in,
                              void* d_out, int out_size, void* d_ws, size_t ws_size,
                              hipStream_t stream) {
  (void)n_in; (void)out_size; (void)ws_size;
  const float* x1     = (const float*)d_in[0];
  const int*   ei1    = (const int*)d_in[1];
  const float* x2     = (const float*)d_in[2];
  const int*   ei2    = (const int*)d_in[3];
  const int*   gidx   = (const int*)d_in[4];
  const float* l1_W1  = (const float*)d_in[5];
  const float* l1_as1 = (const float*)d_in[6];
  const float* l1_ad1 = (const float*)d_in[7];
  const float* l1_b1  = (const float*)d_in[8];
  const float* l1_W2  = (const float*)d_in[9];
  const float* l1_as2 = (const float*)d_in[10];
  const float* l1_ad2 = (const float*)d_in[11];
  const float* l1_b2  = (const float*)d_in[12];
  const float* l1_fcW = (const float*)d_in[13];
  const float* l1_fcb = (const float*)d_in[14];
  const float* in_ln_g = (const float*)d_in[15];
  const float* in_ln_b = (const float*)d_in[16];
  const float* l2_W1  = (const float*)d_in[17];
  const float* l2_as1 = (const float*)d_in[18];
  const float* l2_ad1 = (const float*)d_in[19];
  const float* l2_b1  = (const float*)d_in[20];
  const float* l2_W2  = (const float*)d_in[21];
  const float* l2_as2 = (const float*)d_in[22];
  const float* l2_ad2 = (const float*)d_in[23];
  const float* l2_b2  = (const float*)d_in[24];
  const float* l2_fcW = (const float*)d_in[25];
  const float* l2_fcb = (const float*)d_in[26];
  const float* out_ln_g = (const float*)d_in[27];
  const float* out_ln_b = (const float*)d_in[28];

  const int F_IN = 128, D1 = 128, D2 = 64;
  const long N  = in_sizes[0] / F_IN;
  const long E  = in_sizes[1] / 2;
  const long G  = in_sizes[2] / F_IN;
  const long EG = in_sizes[3] / 2;

  // workspace layout (floats)
  float* ws = (float*)d_ws;
  size_t o = 0;
  float* A    = ws + o; o += (size_t)N * D1;   // ping
  float* Bb   = ws + o; o += (size_t)N * D1;   // pong
  float* asrc = ws + o; o += (size_t)N * 4;
  float* adst = ws + o; o += (size_t)N * 4;
  float* zbuf = ws + o; o += (size_t)N * 4;
  float* gh   = ws + o; o += (size_t)G * D1;
  float* g1   = ws + o; o += (size_t)G * D1;
  float* gasrc = ws + o; o += (size_t)G * 4;
  float* gadst = ws + o; o += (size_t)G * 4;
  float* gz    = ws + o; o += (size_t)G * 4;
  float* adjb  = ws + o; o += (size_t)G * D1;
  float* cnt   = ws + o; o += (size_t)G;
  float* gh2   = ws + o; o += (size_t)G * D2;
  float* g2    = ws + o; o += (size_t)G * D2;
  float* adj2  = ws + o; o += (size_t)G * D2;
  float* Bc = Bb + (size_t)N * D2;             // second half of pong (layer 2)
  float* outp = (float*)d_out;

  auto gemm = [&](const float* Am, const float* Bm, const float* bias,
                  float* Cm, long M, int Nc, int K) {
    long waves = ((M + 15) / 16) * (long)(Nc / 16);
    k_gemm_wmma<<<grid1(waves * 32), 256, 0, stream>>>(Am, Bm, bias, Cm, (int)M, Nc, K);
  };

  // ---------------- Layer 1: node GAT ----------------
  gemm(x1, l1_W1, nullptr, A, N, D1, F_IN);                               // h = x1@W1
  k_attn_coef<<<grid1(N * 4), 256, 0, stream>>>(A, l1_as1, l1_ad1, asrc, adst, N, 4, 32);
  k_fill_bias<<<grid1(N * D1), 256, 0, stream>>>(Bb, l1_b1, N, D1);
  k_fill_zero<<<grid1(N * 4), 256, 0, stream>>>(zbuf, N * 4);
  k_edge_denom<<<grid1((E + N) * 4), 256, 0, stream>>>(ei1, E, N, 4, asrc, adst, zbuf);
  k_edge_aggregate<4, 32, 4><<<grid1((E + N) * 32), 256, 0, stream>>>(
      ei1, E, N, A, asrc, adst, zbuf, Bb);

  // ---------------- Layer 1: group GAT ----------------
  gemm(x2, l1_W2, nullptr, gh, G, D1, F_IN);
  k_attn_coef<<<grid1(G * 4), 256, 0, stream>>>(gh, l1_as2, l1_ad2, gasrc, gadst, G, 4, 32);
  k_fill_bias<<<grid1(G * D1), 256, 0, stream>>>(g1, l1_b2, G, D1);
  k_fill_zero<<<grid1(G * 4), 256, 0, stream>>>(gz, G * 4);
  k_edge_denom<<<grid1((EG + G) * 4), 256, 0, stream>>>(ei2, EG, G, 4, gasrc, gadst, gz);
  k_edge_aggregate<4, 32, 4><<<grid1((EG + G) * 32), 256, 0, stream>>>(
      ei2, EG, G, gh, gasrc, gadst, gz, g1);

  // ---------------- Layer 1: dual fuse + fc + LN + ELU ----------------
  k_fill_zero<<<grid1(G * D1), 256, 0, stream>>>(adjb, G * D1);
  k_fill_zero<<<grid1(G), 256, 0, stream>>>(cnt, G);
  k_adj_accum<<<grid1(EG * D1), 256, 0, stream>>>(ei2, EG, g1, adjb, cnt, D1);
  k_adj_finalize<<<grid1(G * D1), 256, 0, stream>>>(adjb, cnt, G, D1);
  k_fuse<4><<<grid1(N * 32), 256, 0, stream>>>(Bb, g1, adjb, gidx, A, N);   // fused -> A
  gemm(A, l1_fcW, l1_fcb, Bb, N, D1, D1);                                   // fc -> Bb
  k_layernorm<4><<<grid1(N * 32), 256, 0, stream>>>(Bb, in_ln_g, in_ln_b, A, N, 1); // h1f -> A

  // ---------------- Layer 2: node GAT (H=1, C=64) ----------------
  gemm(A, l2_W1, nullptr, Bb, N, D2, D1);                                   // h2 -> Bb[0..N*64)
  k_attn_coef<<<grid1(N), 256, 0, stream>>>(Bb, l2_as1, l2_ad1, asrc, adst, N, 1, 64);
  k_fill_bias<<<grid1(N * D2), 256, 0, stream>>>(Bc, l2_b1, N, D2);
  k_fill_zero<<<grid1(N), 256, 0, stream>>>(zbuf, N);
  k_edge_denom<<<grid1(E + N), 256, 0, stream>>>(ei1, E, N, 1, asrc, adst, zbuf);
  k_edge_aggregate<2, 64, 1><<<grid1((E + N) * 32), 256, 0, stream>>>(
      ei1, E, N, Bb, asrc, adst, zbuf, Bc);

  // ---------------- Layer 2: group GAT ----------------
  gemm(g1, l2_W2, nullptr, gh2, G, D2, D1);
  k_attn_coef<<<grid1(G), 256, 0, stream>>>(gh2, l2_as2, l2_ad2, gasrc, gadst, G, 1, 64);
  k_fill_bias<<<grid1(G * D2), 256, 0, stream>>>(g2, l2_b2, G, D2);
  k_fill_zero<<<grid1(G), 256, 0, stream>>>(gz, G);
  k_edge_denom<<<grid1(EG + G), 256, 0, stream>>>(ei2, EG, G, 1, gasrc, gadst, gz);
  k_edge_aggregate<2, 64, 1><<<grid1((EG + G) * 32), 256, 0, stream>>>(
      ei2, EG, G, gh2, gasrc, gadst, gz, g2);

  // ---------------- Layer 2: dual fuse + fc + LN ----------------
  k_fill_zero<<<grid1(G * D2), 256, 0, stream>>>(adj2, G * D2);
  k_fill_zero<<<grid1(G), 256, 0, stream>>>(cnt, G);
  k_adj_accum<<<grid1(EG * D2), 256, 0, stream>>>(ei2, EG, g2, adj2, cnt, D2);
  k_adj_finalize<<<grid1(G * D2), 256, 0, stream>>>(adj2, cnt, G, D2);
  k_fuse<2><<<grid1(N * 32), 256, 0, stream>>>(Bc, g2, adj2, gidx, A, N);   // fused2 -> A[0..N*64)
  gemm(A, l2_fcW, l2_fcb, Bb, N, D2, D2);                                   // fc -> Bb[0..N*64)
  k_layernorm<2><<<grid1(N * 32), 256, 0, stream>>>(Bb, out_ln_g, out_ln_b, outp, N, 0);

  // ---------------- outputs: (h2 [N,64], g2 [G,64]) ----------------
  k_copy<<<grid1(G * D2), 256, 0, stream>>>(g2, outp + (size_t)N * D2, G * D2);
}